// _InputWeightObserver_63660005261670
// MI455X (gfx1250) — compile-verified
//
#include <hip/hip_runtime.h>
#include <cstdint>

#define COLS 4096
#define CH   4      // rows staged per async chunk (16 KB per chunk per block)
#define NBUF 3      // triple buffer: prefetch distance 2
#define TPB  256

// ---- CDNA5 async global->LDS copy (ASYNCcnt-tracked DMA path) ----
template <bool NT>
__device__ __forceinline__ void async_ld_b128(unsigned lds_off, const float* gptr) {
    if constexpr (NT) {
        // single-use 1 GiB stream: don't pollute the 192 MB L2
        asm volatile("global_load_async_to_lds_b128 %0, %1, off th:TH_LOAD_NT"
                     :: "v"(lds_off), "v"((unsigned long long)(uintptr_t)gptr)
                     : "memory");
    } else {
        asm volatile("global_load_async_to_lds_b128 %0, %1, off"
                     :: "v"(lds_off), "v"((unsigned long long)(uintptr_t)gptr)
                     : "memory");
    }
}

// Single-hardware-instruction float min/max atomics via order-preserving int bits.
// Non-negative floats order like signed ints; negative floats order reversed as
// unsigned ints. The sign-split pair is race-safe (see analysis).
__device__ __forceinline__ void atomicMinF(float* addr, float val) {
    if (val >= 0.0f) atomicMin((int*)addr, __float_as_int(val));
    else             atomicMax((unsigned int*)addr, __float_as_uint(val));
}
__device__ __forceinline__ void atomicMaxF(float* addr, float val) {
    if (val >= 0.0f) atomicMax((int*)addr, __float_as_int(val));
    else             atomicMin((unsigned int*)addr, __float_as_uint(val));
}

__global__ void __launch_bounds__(TPB)
init_kernel(float* __restrict__ out) {
    int i = (int)blockIdx.x * TPB + (int)threadIdx.x;   // 16384 floats
    // tiles of 4096: [x_min]=+inf, [x_max]=-inf, [w_col_min]=+inf, [w_col_max]=-inf
    out[i] = ((i >> 12) & 1) ? -__builtin_inff() : __builtin_inff();
}

// Per-column min/max over `rowsPerBlock` rows of a [*,4096] fp32 matrix.
// grid.x = 4 column tiles (1024 cols each), grid.y = row slices.
// Rows stream through a triple-buffered LDS stage (prefetch distance 2)
// via async b128 DMA; each lane consumes exactly the b128 slot it staged,
// so only per-wave ASYNCcnt/DScnt waits are needed (no workgroup barrier).
template <bool NT>
__global__ void __launch_bounds__(TPB)
col_minmax_kernel(const float* __restrict__ src, int rowsPerBlock,
                  float* __restrict__ out_min, float* __restrict__ out_max) {
    __shared__ __align__(16) float smem[NBUF][CH][1024];   // 48 KB

    const int    t     = (int)threadIdx.x;
    const int    col0  = (int)blockIdx.x * 1024 + t * 4;
    const size_t row0  = (size_t)blockIdx.y * (size_t)rowsPerBlock;
    const float* gbase = src + row0 * COLS + col0;

    // low 32 bits of a generic shared pointer == LDS byte offset
    unsigned ldsSlot[NBUF];
#pragma unroll
    for (int b = 0; b < NBUF; ++b)
        ldsSlot[b] = (unsigned)(uintptr_t)&smem[b][0][t * 4];
    const unsigned rowStrideLds = 1024u * 4u;

    float4 mn = make_float4( __builtin_inff(),  __builtin_inff(),
                             __builtin_inff(),  __builtin_inff());
    float4 mx = make_float4(-__builtin_inff(), -__builtin_inff(),
                            -__builtin_inff(), -__builtin_inff());

    const int nchunks = rowsPerBlock / CH;

    // prologue: stage chunks 0 and 1
#pragma unroll
    for (int r = 0; r < CH; ++r)
        async_ld_b128<NT>(ldsSlot[0] + (unsigned)r * rowStrideLds,
                          gbase + (size_t)r * COLS);
    if (nchunks > 1) {
#pragma unroll
        for (int r = 0; r < CH; ++r)
            async_ld_b128<NT>(ldsSlot[1] + (unsigned)r * rowStrideLds,
                              gbase + (size_t)(CH + r) * COLS);
    }

    int bc = 0;   // buffer holding chunk c
    int bp = 2 % NBUF;   // buffer for chunk c+2
    for (int c = 0; c < nchunks; ++c) {
        if (c + 2 < nchunks) {
            // WAR guard: ds reads of buffer bp (chunk c-1) must retire first
            asm volatile("s_wait_dscnt 0" ::: "memory");
            const float* gnext = gbase + (size_t)(c + 2) * CH * COLS;
#pragma unroll
            for (int r = 0; r < CH; ++r)
                async_ld_b128<NT>(ldsSlot[bp] + (unsigned)r * rowStrideLds,
                                  gnext + (size_t)r * COLS);
            asm volatile("s_wait_asynccnt %0" :: "n"(2 * CH) : "memory"); // chunk c done
        } else if (c + 1 < nchunks) {
            asm volatile("s_wait_asynccnt %0" :: "n"(CH) : "memory");
        } else {
            asm volatile("s_wait_asynccnt 0" ::: "memory");
        }
        // consume chunk c
#pragma unroll
        for (int r = 0; r < CH; ++r) {
            float4 v = *(const float4*)&smem[bc][r][t * 4];
            mn.x = fminf(mn.x, v.x); mn.y = fminf(mn.y, v.y);
            mn.z = fminf(mn.z, v.z); mn.w = fminf(mn.w, v.w);
            mx.x = fmaxf(mx.x, v.x); mx.y = fmaxf(mx.y, v.y);
            mx.z = fmaxf(mx.z, v.z); mx.w = fmaxf(mx.w, v.w);
        }
        bc = (bc + 1 == NBUF) ? 0 : bc + 1;
        bp = (bp + 1 == NBUF) ? 0 : bp + 1;
    }

    // columns are owned 1:1 by (block-tile, thread, component): combine slices
    atomicMinF(&out_min[col0 + 0], mn.x); atomicMinF(&out_min[col0 + 1], mn.y);
    atomicMinF(&out_min[col0 + 2], mn.z); atomicMinF(&out_min[col0 + 3], mn.w);
    atomicMaxF(&out_max[col0 + 0], mx.x); atomicMaxF(&out_max[col0 + 1], mx.y);
    atomicMaxF(&out_max[col0 + 2], mx.z); atomicMaxF(&out_max[col0 + 3], mx.w);
}

// One wave32 per row of w: row min/max + first-occurrence argmin/argmax.
// Runs after the w column pass, so w is L2-resident (64 MB < 192 MB L2).
__global__ void __launch_bounds__(TPB)
row_stats_kernel(const float* __restrict__ w,
                 float* __restrict__ row_min, float* __restrict__ row_max,
                 long long* __restrict__ min_ind, long long* __restrict__ max_ind) {
    const int lane = (int)(threadIdx.x & 31u);
    const int wid  = (int)(threadIdx.x >> 5);
    const int row  = (int)blockIdx.x * 8 + wid;
    const float* p = w + (size_t)row * COLS;

    float bmin =  __builtin_inff(); int bmin_i = 0;
    float bmax = -__builtin_inff(); int bmax_i = 0;

#pragma unroll 8
    for (int it = 0; it < 32; ++it) {
        int col = it * 128 + lane * 4;          // coalesced b128 across the wave
        float4 v = *(const float4*)(p + col);
        if (v.x < bmin) { bmin = v.x; bmin_i = col;     }
        if (v.y < bmin) { bmin = v.y; bmin_i = col + 1; }
        if (v.z < bmin) { bmin = v.z; bmin_i = col + 2; }
        if (v.w < bmin) { bmin = v.w; bmin_i = col + 3; }
        if (v.x > bmax) { bmax = v.x; bmax_i = col;     }
        if (v.y > bmax) { bmax = v.y; bmax_i = col + 1; }
        if (v.z > bmax) { bmax = v.z; bmax_i = col + 2; }
        if (v.w > bmax) { bmax = v.w; bmax_i = col + 3; }
    }

    // wave32 butterfly reduction with first-occurrence (lowest index) tie-break
    for (int off = 16; off > 0; off >>= 1) {
        float ov = __shfl_xor(bmin,   off, 32);
        int   oi = __shfl_xor(bmin_i, off, 32);
        if (ov < bmin || (ov == bmin && oi < bmin_i)) { bmin = ov; bmin_i = oi; }
        float pv = __shfl_xor(bmax,   off, 32);
        int   pi = __shfl_xor(bmax_i, off, 32);
        if (pv > bmax || (pv == bmax && pi < bmax_i)) { bmax = pv; bmax_i = pi; }
    }

    if (lane == 0) {
        row_min[row] = bmin;
        row_max[row] = bmax;
        min_ind[row] = (long long)bmin_i;
        max_ind[row] = (long long)bmax_i;
    }
}

extern "C" void kernel_launch(void* const* d_in, const int* in_sizes, int n_in,
                              void* d_out, int out_size, void* d_ws, size_t ws_size,
                              hipStream_t stream) {
    const float* x = (const float*)d_in[0];   // [65536, 4096]
    const float* w = (const float*)d_in[1];   // [4096, 4096]

    float* out       = (float*)d_out;
    float* x_min     = out;                   // [4096]
    float* x_max     = out + 4096;            // [4096]
    float* w_col_min = out + 8192;            // [4096]
    float* w_col_max = out + 12288;           // [4096]
    float* w_row_min = out + 16384;           // [4096]
    float* w_row_max = out + 20480;           // [4096]
    long long* min_ind = (long long*)(out + 24576);   // [4096] int64
    long long* max_ind = min_ind + 4096;              // [4096] int64

    // 1) seed atomic-reduced outputs with identities (deterministic each call)
    init_kernel<<<64, TPB, 0, stream>>>(out);

    // 2) x per-column min/max: 4 col-tiles x 128 row-slices (512 rows each),
    //    NT streaming (don't cache 1 GiB in L2)
    col_minmax_kernel<true><<<dim3(4, 128), TPB, 0, stream>>>(x, 512, x_min, x_max);

    // 3) w per-column min/max: RT loads so w lands in L2 for the row pass
    col_minmax_kernel<false><<<dim3(4, 32), TPB, 0, stream>>>(w, 128, w_col_min, w_col_max);

    // 4) w per-row min/max + first-occurrence argmin/argmax (1 wave / row, L2 hits)
    row_stats_kernel<<<512, TPB, 0, stream>>>(w, w_row_min, w_row_max, min_ind, max_ind);
}